// GraphConv_19610820674287
// MI455X (gfx1250) — compile-verified
//
#include <hip/hip_runtime.h>

// ---------------------------------------------------------------------------
// GraphConv / NNConv+GRU message passing for MI455X (gfx1250, wave32)
//   - edge embedding GEMM  : v_wmma_f32_16x16x4_f32, K=128, explicit
//                            double-buffered pipeline (no fragment copies)
//   - BatchNorm            : folded into per-column affine a*e+b
//   - per-edge msg GEMV    : wave-per-edge, b64 loads, acoef in LDS,
//                            bias term precomputed via WMMA (xb = x @ Bcoef)
//   - GRU GEMMs            : v_wmma_f32_16x16x4_f32, fused gate epilogue
// ---------------------------------------------------------------------------

#define NNODE 50000
#define NEDGE 100000
#define DD    64
#define DSQ   4096          // D*D
#define EPSBN 1e-5f

typedef __attribute__((ext_vector_type(2))) float v2f;
typedef __attribute__((ext_vector_type(8))) float v8f;

__device__ __forceinline__ v8f splat8(float x) {
    v8f r;
#pragma unroll
    for (int i = 0; i < 8; ++i) r[i] = x;
    return r;
}
__device__ __forceinline__ float sigm(float x) { return 1.0f / (1.0f + __expf(-x)); }

// ---------------------------------------------------------------------------
// 0) transpose W_edge [128,4096] -> Wt [4096,128]  (one-time, 2MB)
// ---------------------------------------------------------------------------
__global__ void k_transpose(const float* __restrict__ We, float* __restrict__ Wt) {
    int i = blockIdx.x * 256 + threadIdx.x;          // coalesced read of We
    if (i >= 128 * DSQ) return;
    int k = i >> 12, c = i & (DSQ - 1);
    Wt[(size_t)c * 128 + k] = We[i];
}

// ---------------------------------------------------------------------------
// 1) e_raw[E,4096] = edge[E,128] @ W_edge[128,4096]   (f32 WMMA)
//    wave = 16 rows x 128 cols (8 C tiles), K-loop 32 x (16x16x4),
//    explicit double-buffered fragments: sets A and B alternate, k0 += 8
// ---------------------------------------------------------------------------
__global__ __launch_bounds__(256) void k_edge_gemm(const float* __restrict__ edge,
                                                   const float* __restrict__ Wt,
                                                   float* __restrict__ eraw) {
    const int lane  = threadIdx.x & 31;
    const int wave  = blockIdx.x * 8 + (threadIdx.x >> 5);
    const int rowT  = wave >> 5;          // 6250 row tiles
    const int colG  = wave & 31;          // 32 col groups of 128
    const int r0    = rowT * 16;
    const int c0    = colG * 128;
    const int n     = lane & 15;
    const int khalf = (lane >> 4) * 2;    // lanes 16-31 carry K+2,K+3

    const float* __restrict__ arow  = edge + (size_t)(r0 + n) * 128 + khalf;
    const float* __restrict__ wbase = Wt + (size_t)(c0 + n) * 128 + khalf;

    v8f acc[8];
#pragma unroll
    for (int t = 0; t < 8; ++t) acc[t] = splat8(0.0f);

    // set A <- k = 0
    v2f aA = *(const v2f*)(arow);
    v2f bA0 = *(const v2f*)(wbase);
    v2f bA1 = *(const v2f*)(wbase + 1 * 2048);
    v2f bA2 = *(const v2f*)(wbase + 2 * 2048);
    v2f bA3 = *(const v2f*)(wbase + 3 * 2048);
    v2f bA4 = *(const v2f*)(wbase + 4 * 2048);
    v2f bA5 = *(const v2f*)(wbase + 5 * 2048);
    v2f bA6 = *(const v2f*)(wbase + 6 * 2048);
    v2f bA7 = *(const v2f*)(wbase + 7 * 2048);
    v2f aB, bB0, bB1, bB2, bB3, bB4, bB5, bB6, bB7;

#define WMMA8(AF, B0, B1, B2, B3, B4, B5, B6, B7)                                          \
    acc[0] = __builtin_amdgcn_wmma_f32_16x16x4_f32(false, AF, false, B0, (short)0, acc[0], false, false); \
    acc[1] = __builtin_amdgcn_wmma_f32_16x16x4_f32(false, AF, false, B1, (short)0, acc[1], false, false); \
    acc[2] = __builtin_amdgcn_wmma_f32_16x16x4_f32(false, AF, false, B2, (short)0, acc[2], false, false); \
    acc[3] = __builtin_amdgcn_wmma_f32_16x16x4_f32(false, AF, false, B3, (short)0, acc[3], false, false); \
    acc[4] = __builtin_amdgcn_wmma_f32_16x16x4_f32(false, AF, false, B4, (short)0, acc[4], false, false); \
    acc[5] = __builtin_amdgcn_wmma_f32_16x16x4_f32(false, AF, false, B5, (short)0, acc[5], false, false); \
    acc[6] = __builtin_amdgcn_wmma_f32_16x16x4_f32(false, AF, false, B6, (short)0, acc[6], false, false); \
    acc[7] = __builtin_amdgcn_wmma_f32_16x16x4_f32(false, AF, false, B7, (short)0, acc[7], false, false)

#define LOADSET(AF, B0, B1, B2, B3, B4, B5, B6, B7, KOFF)                                  \
    AF = *(const v2f*)(arow + (KOFF));                                                     \
    B0 = *(const v2f*)(wbase + 0 * 2048 + (KOFF));                                         \
    B1 = *(const v2f*)(wbase + 1 * 2048 + (KOFF));                                         \
    B2 = *(const v2f*)(wbase + 2 * 2048 + (KOFF));                                         \
    B3 = *(const v2f*)(wbase + 3 * 2048 + (KOFF));                                         \
    B4 = *(const v2f*)(wbase + 4 * 2048 + (KOFF));                                         \
    B5 = *(const v2f*)(wbase + 5 * 2048 + (KOFF));                                         \
    B6 = *(const v2f*)(wbase + 6 * 2048 + (KOFF));                                         \
    B7 = *(const v2f*)(wbase + 7 * 2048 + (KOFF))

#pragma unroll 1
    for (int k0 = 0; k0 < 120; k0 += 8) {
        LOADSET(aB, bB0, bB1, bB2, bB3, bB4, bB5, bB6, bB7, k0 + 4);
        WMMA8(aA, bA0, bA1, bA2, bA3, bA4, bA5, bA6, bA7);        // k-step k0
        LOADSET(aA, bA0, bA1, bA2, bA3, bA4, bA5, bA6, bA7, k0 + 8);
        WMMA8(aB, bB0, bB1, bB2, bB3, bB4, bB5, bB6, bB7);        // k-step k0+4
    }
    // epilogue: k = 120 (in set A) and k = 124
    LOADSET(aB, bB0, bB1, bB2, bB3, bB4, bB5, bB6, bB7, 124);
    WMMA8(aA, bA0, bA1, bA2, bA3, bA4, bA5, bA6, bA7);
    WMMA8(aB, bB0, bB1, bB2, bB3, bB4, bB5, bB6, bB7);

#undef WMMA8
#undef LOADSET

    const int rbase = r0 + (lane >> 4) * 8;   // C layout: VGPR v -> row v (+8)
#pragma unroll
    for (int t = 0; t < 8; ++t) {
        const int c = c0 + t * 16 + n;
#pragma unroll
        for (int v = 0; v < 8; ++v)
            eraw[(size_t)(rbase + v) * DSQ + c] = acc[t][v];
    }
}

// ---------------------------------------------------------------------------
// 2) BN stats: per-column sum / sumsq over E rows (float4 streaming pass)
// ---------------------------------------------------------------------------
#define ROWCHUNK 625
__global__ __launch_bounds__(256) void k_bn_stats(const float* __restrict__ eraw,
                                                  float* __restrict__ sums,
                                                  float* __restrict__ sumsq) {
    const int c4 = (blockIdx.x * 256 + threadIdx.x) * 4;  // 4096 cols / 4
    const int r0 = blockIdx.y * ROWCHUNK;
    float s0 = 0.f, s1 = 0.f, s2 = 0.f, s3 = 0.f;
    float q0 = 0.f, q1 = 0.f, q2 = 0.f, q3 = 0.f;
    for (int r = r0; r < r0 + ROWCHUNK; ++r) {
        float4 v = *(const float4*)&eraw[(size_t)r * DSQ + c4];
        s0 += v.x; s1 += v.y; s2 += v.z; s3 += v.w;
        q0 += v.x * v.x; q1 += v.y * v.y; q2 += v.z * v.z; q3 += v.w * v.w;
    }
    atomicAdd(&sums[c4 + 0], s0);  atomicAdd(&sums[c4 + 1], s1);
    atomicAdd(&sums[c4 + 2], s2);  atomicAdd(&sums[c4 + 3], s3);
    atomicAdd(&sumsq[c4 + 0], q0); atomicAdd(&sumsq[c4 + 1], q1);
    atomicAdd(&sumsq[c4 + 2], q2); atomicAdd(&sumsq[c4 + 3], q3);
}

__global__ void k_bn_finalize(const float* __restrict__ sums,
                              const float* __restrict__ sumsq,
                              const float* __restrict__ gamma,
                              const float* __restrict__ beta,
                              float* __restrict__ acoef,
                              float* __restrict__ bcoef) {
    const int c = blockIdx.x * 256 + threadIdx.x;
    if (c >= DSQ) return;
    const float inv = 1.0f / (float)NEDGE;
    float mu  = sums[c] * inv;
    float var = fmaxf(sumsq[c] * inv - mu * mu, 0.0f);
    float a   = gamma[c] * rsqrtf(var + EPSBN);
    acoef[c]  = a;
    bcoef[c]  = beta[c] - a * mu;
}

// ---------------------------------------------------------------------------
// utility kernels (float4 vectorized)
// ---------------------------------------------------------------------------
__global__ void k_fill4(float4* __restrict__ p, int n4) {
    int i = blockIdx.x * 256 + threadIdx.x;
    if (i < n4) p[i] = make_float4(0.f, 0.f, 0.f, 0.f);
}
__global__ void k_copy4(float4* __restrict__ d, const float4* __restrict__ s, int n4) {
    int i = blockIdx.x * 256 + threadIdx.x;
    if (i < n4) d[i] = s[i];
}
__global__ void k_count(const long long* __restrict__ dst, float* __restrict__ cnt) {
    int e = blockIdx.x * 256 + threadIdx.x;
    if (e < NEDGE) atomicAdd(&cnt[(int)dst[e]], 1.0f);
}
__global__ void k_accum4(float4* __restrict__ accum, const float4* __restrict__ node,
                         const float4* __restrict__ x0, int n4) {
    int i = blockIdx.x * 256 + threadIdx.x;
    if (i >= n4) return;
    float4 a = accum[i], b = node[i], c = x0[i];
    a.x += b.x + c.x; a.y += b.y + c.y; a.z += b.z + c.z; a.w += b.w + c.w;
    accum[i] = a;
}
__global__ void k_final4(float4* __restrict__ out, const float4* __restrict__ accum, int n4) {
    int i = blockIdx.x * 256 + threadIdx.x;
    if (i >= n4) return;
    float4 a = accum[i];
    out[i] = make_float4(a.x * 0.25f, a.y * 0.25f, a.z * 0.25f, a.w * 0.25f);
}
// node_in = relu(agg / max(cnt,1) + bias)
__global__ void k_node_in(const float4* __restrict__ agg, const float* __restrict__ cnt,
                          const float* __restrict__ bias, float4* __restrict__ xin) {
    int i = blockIdx.x * 256 + threadIdx.x;      // over ND/4
    if (i >= NNODE * DD / 4) return;
    int row = i >> 4;
    float inv = 1.0f / fmaxf(cnt[row], 1.0f);
    float4 a = agg[i];
    float4 b = *(const float4*)&bias[(i & 15) * 4];
    xin[i] = make_float4(fmaxf(a.x * inv + b.x, 0.f), fmaxf(a.y * inv + b.y, 0.f),
                         fmaxf(a.z * inv + b.z, 0.f), fmaxf(a.w * inv + b.w, 0.f));
}

// ---------------------------------------------------------------------------
// 3a) xb[N,64] = node @ Bcoef[64,64]  (WMMA) — bias part of BN-folded message
// ---------------------------------------------------------------------------
__global__ __launch_bounds__(128) void k_xb(const float* __restrict__ node_cur,
                                            const float* __restrict__ bcoef,
                                            float* __restrict__ xb) {
    const int lane  = threadIdx.x & 31;
    const int ct    = threadIdx.x >> 5;
    const int r0    = blockIdx.x * 16;
    const int n     = lane & 15;
    const int col   = ct * 16 + n;
    const int khalf = (lane >> 4) * 2;
    const int mrow  = r0 + n;

    v8f acc = splat8(0.0f);
    for (int k0 = 0; k0 < DD; k0 += 4) {
        const int kb = k0 + khalf;
        v2f a = *(const v2f*)&node_cur[mrow * DD + kb];
        v2f b;
        b.x = bcoef[kb * DD + col];
        b.y = bcoef[(kb + 1) * DD + col];
        acc = __builtin_amdgcn_wmma_f32_16x16x4_f32(false, a, false, b, (short)0, acc, false, false);
    }
    const int rbase = r0 + (lane >> 4) * 8;
#pragma unroll
    for (int v = 0; v < 8; ++v)
        xb[(rbase + v) * DD + col] = acc[v];
}

// ---------------------------------------------------------------------------
// 3b) per-edge message: agg[dst] += xb[src] + sum_d x_src[d]*a[d,k]*e_raw[e,d,k]
//     wave-per-edge; lane owns 2 adjacent cols -> b64 loads; acoef in LDS
// ---------------------------------------------------------------------------
__global__ __launch_bounds__(256) void k_msg(const long long* __restrict__ src,
                                             const long long* __restrict__ dst,
                                             const float* __restrict__ node,
                                             const float* __restrict__ eraw,
                                             const float* __restrict__ acoef,
                                             const float* __restrict__ xb,
                                             float* __restrict__ agg) {
    __shared__ float sA[DSQ];                       // 16 KB of 320 KB WGP LDS
    for (int i = threadIdx.x; i < DSQ; i += 256) sA[i] = acoef[i];
    __syncthreads();

    const int lane = threadIdx.x & 31;
    const int e    = blockIdx.x * 8 + (threadIdx.x >> 5);
    if (e >= NEDGE) return;
    const int s = (int)src[e];
    const int t = (int)dst[e];
    const float xlo = node[s * DD + lane];
    const float xhi = node[s * DD + 32 + lane];
    const size_t base = (size_t)e * DSQ;
    const int cl = 2 * lane;                        // this lane's 2 columns
    float acc0 = 0.f, acc1 = 0.f;
#pragma unroll 4
    for (int d = 0; d < DD; ++d) {
        const float xs = (d < 32) ? xlo : xhi;
        const float xd = __shfl(xs, d & 31, 32);
        v2f ev = *(const v2f*)&eraw[base + d * DD + cl];
        v2f av = *(const v2f*)&sA[d * DD + cl];
        acc0 = fmaf(xd * av.x, ev.x, acc0);
        acc1 = fmaf(xd * av.y, ev.y, acc1);
    }
    v2f xbv = *(const v2f*)&xb[s * DD + cl];
    atomicAdd(&agg[t * DD + cl],     acc0 + xbv.x);
    atomicAdd(&agg[t * DD + cl + 1], acc1 + xbv.y);
}

// ---------------------------------------------------------------------------
// 4) fused GRU step (WMMA): per wave one 16-row x 16-col tile.
// ---------------------------------------------------------------------------
__global__ __launch_bounds__(128) void k_gru(const float* __restrict__ xin,
                                             const float* __restrict__ hcur,
                                             const float* __restrict__ Wih,
                                             const float* __restrict__ Whh,
                                             const float* __restrict__ bih,
                                             const float* __restrict__ bhh,
                                             float* __restrict__ hnext) {
    const int lane  = threadIdx.x & 31;
    const int ct    = threadIdx.x >> 5;       // col tile 0..3
    const int r0    = blockIdx.x * 16;
    const int n     = lane & 15;
    const int col   = ct * 16 + n;
    const int khalf = (lane >> 4) * 2;
    const int mrow  = r0 + n;

    v8f accR = splat8(bih[col]       + bhh[col]);
    v8f accZ = splat8(bih[64 + col]  + bhh[64 + col]);
    v8f accN = splat8(bih[128 + col]);
    v8f accH = splat8(bhh[128 + col]);

    for (int k0 = 0; k0 < DD; k0 += 4) {
        const int kb = k0 + khalf;
        v2f ax = *(const v2f*)&xin[mrow * DD + kb];
        v2f ah = *(const v2f*)&hcur[mrow * DD + kb];
        v2f b;
        b = *(const v2f*)&Wih[col * DD + kb];
        accR = __builtin_amdgcn_wmma_f32_16x16x4_f32(false, ax, false, b, (short)0, accR, false, false);
        b = *(const v2f*)&Whh[col * DD + kb];
        accR = __builtin_amdgcn_wmma_f32_16x16x4_f32(false, ah, false, b, (short)0, accR, false, false);
        b = *(const v2f*)&Wih[(64 + col) * DD + kb];
        accZ = __builtin_amdgcn_wmma_f32_16x16x4_f32(false, ax, false, b, (short)0, accZ, false, false);
        b = *(const v2f*)&Whh[(64 + col) * DD + kb];
        accZ = __builtin_amdgcn_wmma_f32_16x16x4_f32(false, ah, false, b, (short)0, accZ, false, false);
        b = *(const v2f*)&Wih[(128 + col) * DD + kb];
        accN = __builtin_amdgcn_wmma_f32_16x16x4_f32(false, ax, false, b, (short)0, accN, false, false);
        b = *(const v2f*)&Whh[(128 + col) * DD + kb];
        accH = __builtin_amdgcn_wmma_f32_16x16x4_f32(false, ah, false, b, (short)0, accH, false, false);
    }

    const int rbase = r0 + (lane >> 4) * 8;
#pragma unroll
    for (int v = 0; v < 8; ++v) {
        const float r  = sigm(accR[v]);
        const float z  = sigm(accZ[v]);
        const float nn = tanhf(accN[v] + r * accH[v]);
        const float hv = hcur[(rbase + v) * DD + col];
        hnext[(rbase + v) * DD + col] = (1.0f - z) * nn + z * hv;
    }
}

// ---------------------------------------------------------------------------
// host-side launcher
// ---------------------------------------------------------------------------
extern "C" void kernel_launch(void* const* d_in, const int* in_sizes, int n_in,
                              void* d_out, int out_size, void* d_ws, size_t ws_size,
                              hipStream_t stream) {
    const float*     node  = (const float*)d_in[0];
    const long long* ei    = (const long long*)d_in[1];   // [2, E] int64
    const float*     edge  = (const float*)d_in[2];
    const float*     We    = (const float*)d_in[3];
    const float*     gamma = (const float*)d_in[4];
    const float*     beta  = (const float*)d_in[5];
    const float*     bias  = (const float*)d_in[6];
    const float*     Wih   = (const float*)d_in[7];
    const float*     Whh   = (const float*)d_in[8];
    const float*     bih   = (const float*)d_in[9];
    const float*     bhh   = (const float*)d_in[10];
    float*           out   = (float*)d_out;
    (void)in_sizes; (void)n_in; (void)out_size; (void)ws_size;

    const long long* srcI = ei;
    const long long* dstI = ei + NEDGE;

    // workspace layout (256B aligned slabs)
    char* ws = (char*)d_ws;
    size_t off = 0;
    auto take = [&](size_t nfloats) {
        float* p = (float*)(ws + off);
        off += ((nfloats * sizeof(float) + 255) / 256) * 256;
        return p;
    };
    float* eraw  = take((size_t)NEDGE * DSQ);   // 1.64 GB
    float* agg   = take((size_t)NNODE * DD);
    float* xin   = take((size_t)NNODE * DD);
    float* xb    = take((size_t)NNODE * DD);
    float* h0    = take((size_t)NNODE * DD);
    float* h1    = take((size_t)NNODE * DD);
    float* accum = take((size_t)NNODE * DD);
    float* Wt    = take((size_t)128 * DSQ);     // transposed edge weights
    float* sums  = take(DSQ);
    float* sumsq = take(DSQ);
    float* acoef = take(DSQ);
    float* bcoef = take(DSQ);
    float* cnt   = take(NNODE);

    const int ND  = NNODE * DD;        // 3,200,000
    const int ND4 = ND / 4;            // 800,000
    const int gND4 = ND4 / 256;        // 3125 exact

    // weight transpose + edge embedding GEMM
    k_transpose<<<(128 * DSQ) / 256, 256, 0, stream>>>(We, Wt);
    k_edge_gemm<<<25000, 256, 0, stream>>>(edge, Wt, eraw);

    // BN stats + folded affine coefficients
    k_fill4<<<4, 256, 0, stream>>>((float4*)sums, DSQ / 4);
    k_fill4<<<4, 256, 0, stream>>>((float4*)sumsq, DSQ / 4);
    k_bn_stats<<<dim3(4, NEDGE / ROWCHUNK), 256, 0, stream>>>(eraw, sums, sumsq);
    k_bn_finalize<<<DSQ / 256, 256, 0, stream>>>(sums, sumsq, gamma, beta, acoef, bcoef);

    // degree counts + state init
    k_fill4<<<(NNODE / 4 + 255) / 256, 256, 0, stream>>>((float4*)cnt, NNODE / 4);
    k_count<<<(NEDGE + 255) / 256, 256, 0, stream>>>(dstI, cnt);
    k_copy4<<<gND4, 256, 0, stream>>>((float4*)h0, (const float4*)node, ND4);
    k_fill4<<<gND4, 256, 0, stream>>>((float4*)accum, ND4);

    // 4 message-passing + GRU steps (h ping-pong)
    for (int s = 0; s < 4; ++s) {
        float* hc = (s & 1) ? h1 : h0;
        float* hn = (s & 1) ? h0 : h1;
        k_accum4<<<gND4, 256, 0, stream>>>((float4*)accum, (const float4*)hc,
                                           (const float4*)node, ND4);
        k_fill4<<<gND4, 256, 0, stream>>>((float4*)agg, ND4);
        k_xb<<<NNODE / 16, 128, 0, stream>>>(hc, bcoef, xb);
        k_msg<<<NEDGE / 8, 256, 0, stream>>>(srcI, dstI, hc, eraw, acoef, xb, agg);
        k_node_in<<<gND4, 256, 0, stream>>>((const float4*)agg, cnt, bias, (float4*)xin);
        k_gru<<<NNODE / 16, 128, 0, stream>>>(xin, hc, Wih, Whh, bih, bhh, hn);
    }
    k_final4<<<gND4, 256, 0, stream>>>((float4*)out, (const float4*)accum, ND4);
}